// Conv4d_64141041598570
// MI455X (gfx1250) — compile-verified
//
#include <hip/hip_runtime.h>

typedef float v2f __attribute__((ext_vector_type(2)));
typedef float v8f __attribute__((ext_vector_type(8)));

#define S    24
#define S2   576       // 24^2
#define S3   13824     // 24^3
#define S4   331776    // 24^4
#define CS4  5308416   // 16 * 24^4
#define KTOT 1296      // C_IN * 3^4

#define PW   26        // padded plane width (halo 1 each side)
#define PP   676       // 26*26
#define NSB  2704      // 4 channels * 676  (B stage, floats)
#define NSA  576       // 16 rows * 36 klocal (A stage, floats)

// LDS float-offset for klocal = (cl, kh, kw) inside a chunk: cl*676 + kh*26 + kw
__device__ __forceinline__ constexpr int VC(int kl) {
    return (kl / 9) * PP + ((kl % 9) / 3) * PW + ((kl % 9) % 3);
}

// One block = one (n,u,v) slab, full 24x24 (h,w) plane.
// 9 waves; each wave owns 4 WMMA tiles (64 positions). K consumed with
// v_wmma_f32_16x16x4_f32 in 36 chunks of 9 groups; per chunk the input
// plane (4 channels, halo-padded/zero-filled) and the weight slice are
// staged in LDS so the inner loop has no bounds checks and all LDS
// offsets are compile-time constants (khalf-selected once up front).
__global__ __launch_bounds__(288)
void conv4d_wmma_f32(const float* __restrict__ inp,
                     const float* __restrict__ wgt,
                     const float* __restrict__ bias,
                     float* __restrict__ out)
{
    __shared__ __align__(16) float sb[NSB];  // B: [cl][hh][ww]
    __shared__ __align__(16) float sa[NSA];  // A: [m][klocal]

    const int tid = threadIdx.x;
    const int b   = blockIdx.x;
    const int n   = b / S2;
    const int uv  = b - n * S2;
    const int u   = uv / S;
    const int v   = uv - u * S;

    const int lane  = tid & 31;
    const int wave  = tid >> 5;
    const int m     = lane & 15;
    const int khalf = lane >> 4;

    // ---- per-thread B-staging precompute (constant across chunks) ----
    int spo[10];
    int svalid = 0;
    #pragma unroll
    for (int j = 0; j < 10; ++j) {
        int idx = tid + j * 288;
        int cl  = idx / PP;
        int r   = idx - cl * PP;
        int hh  = r / PW - 1;
        int ww  = (r - (r / PW) * PW) - 1;
        spo[j]  = cl * S4 + hh * S + ww;
        if (idx < NSB && (unsigned)hh < (unsigned)S && (unsigned)ww < (unsigned)S)
            svalid |= (1 << j);
    }
    // ---- per-thread A-staging precompute ----
    int adst[2], asrc[2];
    #pragma unroll
    for (int j = 0; j < 2; ++j) {
        int idx = tid + j * 288;         // 0..575
        int mm  = idx / 36;
        int kl  = idx - mm * 36;
        int cl  = kl / 9;
        int r   = kl - cl * 9;
        int kh  = r / 3;
        int kw  = r - kh * 3;
        adst[j] = idx;
        asrc[j] = mm * KTOT + cl * 81 + kh * 3 + kw;   // + chunk offset later
    }

    // ---- B LDS offsets per (group, element), selected once on khalf ----
    int voff[9][2];
    #pragma unroll
    for (int g = 0; g < 9; ++g) {
        #pragma unroll
        for (int e = 0; e < 2; ++e) {
            voff[g][e] = khalf ? VC(4 * g + 2 + e) : VC(4 * g + e);
        }
    }

    // ---- per-tile bases ----
    int vb[4], obase[4];
    #pragma unroll
    for (int t = 0; t < 4; ++t) {
        int pos = wave * 64 + t * 16 + m;   // flattened (h,w)
        int h = pos / S;
        int w = pos - h * S;
        vb[t]    = h * PW + w;              // sb float index base
        obase[t] = n * CS4 + u * S3 + v * S2 + h * S + w;
    }

    v8f acc[4];
    #pragma unroll
    for (int t = 0; t < 4; ++t) acc[t] = (v8f){};

    for (int ku = 0; ku < 3; ++ku) {
        const int ui = u + ku - 1;
        for (int kv = 0; kv < 3; ++kv) {
            const int vi = v + kv - 1;
            // block-uniform: whole (ui,vi) plane out of range -> contributes 0
            if (!(((unsigned)ui < (unsigned)S) & ((unsigned)vi < (unsigned)S)))
                continue;
            for (int cq = 0; cq < 4; ++cq) {
                const int sc = cq * 4 * 81 + ku * 27 + kv * 9; // weight chunk base
                const int cb = n * CS4 + (cq * 4) * S4 + ui * S3 + vi * S2;

                // stage B plane (zero-filled halo)
                #pragma unroll
                for (int j = 0; j < 10; ++j) {
                    int idx = tid + j * 288;
                    if (idx < NSB) {
                        float val = 0.0f;
                        if ((svalid >> j) & 1) val = inp[cb + spo[j]];
                        sb[idx] = val;
                    }
                }
                // stage A slice (performs the K-permutation)
                #pragma unroll
                for (int j = 0; j < 2; ++j)
                    sa[adst[j]] = wgt[asrc[j] + sc];

                __syncthreads();

                const float* sarow = sa + m * 36 + 2 * khalf;
                #pragma unroll
                for (int g = 0; g < 9; ++g) {
                    v2f a = *(const v2f*)(sarow + 4 * g);   // ds_load_b64
                    #pragma unroll
                    for (int t = 0; t < 4; ++t) {
                        v2f bv;
                        bv.x = sb[vb[t] + voff[g][0]];      // ds_load_b32
                        bv.y = sb[vb[t] + voff[g][1]];      // ds_load_b32
                        acc[t] = __builtin_amdgcn_wmma_f32_16x16x4_f32(
                                     false, a, false, bv, (short)0, acc[t],
                                     false, false);
                    }
                }

                __syncthreads();
            }
        }
    }

    // Epilogue: C/D layout — VGPR r: lanes 0-15 -> M=r, lanes 16-31 -> M=r+8.
    #pragma unroll
    for (int r = 0; r < 8; ++r) {
        const int co = (khalf << 3) + r;
        const float bb = bias[co];
        #pragma unroll
        for (int t = 0; t < 4; ++t) {
            out[obase[t] + co * S4] = acc[t][r] + bb;
        }
    }
}

extern "C" void kernel_launch(void* const* d_in, const int* in_sizes, int n_in,
                              void* d_out, int out_size, void* d_ws, size_t ws_size,
                              hipStream_t stream) {
    const float* inp  = (const float*)d_in[0];   // (2,16,24,24,24,24) fp32
    const float* wgt  = (const float*)d_in[1];   // (16, 1296) fp32
    const float* bias = (const float*)d_in[2];   // (16) fp32
    float* out = (float*)d_out;

    (void)in_sizes; (void)n_in; (void)out_size; (void)d_ws; (void)ws_size;

    dim3 grid(2 * S * S);   // 1152 blocks: one per (n,u,v)
    dim3 block(288);        // 9 waves
    conv4d_wmma_f32<<<grid, block, 0, stream>>>(inp, wgt, bias, out);
}